// GNN_29343216566348
// MI455X (gfx1250) — compile-verified
//
#include <hip/hip_runtime.h>
#include <hip/hip_bf16.h>

// ---------------------------------------------------------------------------
// GNN forward: gcn_norm -> 4x SpMM(+I) -> GEMM(64x64)+bias -> BN -> SELU
//              -> SpMM(+I) -> GEMM(64x32)+bias -> softmax(32)
// MI455X notes: feature matrix (25.6MB) is L2-resident (192MB L2), so the
// edge-parallel SpMM is L2-bandwidth bound. Dense GEMMs use fp32 WMMA
// (v_wmma_f32_16x16x4_f32) to hit the matrix pipes without losing precision.
// ---------------------------------------------------------------------------

typedef __attribute__((ext_vector_type(2))) float v2f;
typedef __attribute__((ext_vector_type(8))) float v8f;

__device__ __forceinline__ v8f wmma4(v2f a, v2f b, v8f c) {
  // 8 args: (neg_a, A, neg_b, B, c_mod, C, reuse_a, reuse_b)
  return __builtin_amdgcn_wmma_f32_16x16x4_f32(false, a, false, b, (short)0, c,
                                               false, false);
}

// ---------------- utility kernels ----------------
__global__ void k_zero(float* __restrict__ p, long n) {
  long i = (long)blockIdx.x * blockDim.x + threadIdx.x;
  if (i < n) p[i] = 0.0f;
}

__global__ void k_copy4(const float4* __restrict__ src, float4* __restrict__ dst,
                        long n4) {
  long i = (long)blockIdx.x * blockDim.x + threadIdx.x;
  if (i < n4) dst[i] = src[i];
}

// ---------------- gcn_norm ----------------
__global__ void k_deg(const int* __restrict__ row, float* __restrict__ deg, int E) {
  int e = blockIdx.x * blockDim.x + threadIdx.x;
  if (e < E) atomicAdd(&deg[row[e]], 1.0f);
}

__global__ void k_dis(const float* __restrict__ deg, float* __restrict__ dis, int n) {
  int i = blockIdx.x * blockDim.x + threadIdx.x;
  if (i < n) {
    float d = deg[i];
    dis[i] = (d > 0.0f) ? rsqrtf(fmaxf(d, 1.0f)) : 0.0f;
  }
}

__global__ void k_edgew(const int* __restrict__ row, const int* __restrict__ col,
                        const float* __restrict__ dis, float* __restrict__ w, int E) {
  int e = blockIdx.x * blockDim.x + threadIdx.x;
  if (e < E) w[e] = dis[row[e]] * dis[col[e]];
}

// ---------------- SpMM accumulate: xout[row[e],:] += w[e]*xin[col[e],:] ------
// 256 threads = 16 edges x 16 threads; each thread does a float4 (16B) slice.
// Gathers and atomics land in L2 (feature matrix is L2-resident on MI455X).
__global__ void k_spmm(const int* __restrict__ row, const int* __restrict__ col,
                       const float* __restrict__ w, const float* __restrict__ xin,
                       float* __restrict__ xout, int E) {
  int t = threadIdx.x;
  int e = blockIdx.x * 16 + (t >> 4);
  if (e >= E) return;
  int f4 = (t & 15) * 4;
  int r = row[e], c = col[e];
  float we = w[e];
  float4 v = *(const float4*)(xin + (size_t)c * 64 + f4);
  float* dst = xout + (size_t)r * 64 + f4;
  atomicAdd(dst + 0, we * v.x);
  atomicAdd(dst + 1, we * v.y);
  atomicAdd(dst + 2, we * v.z);
  atomicAdd(dst + 3, we * v.w);
}

// ---------------- dense GEMM via fp32 WMMA ----------------
// H[n x NOUT] = X[n x 64] @ W[64 x NOUT] + bias. One wave per 16-row tile,
// 16 K-steps of v_wmma_f32_16x16x4_f32, A-fragment reused across N-tiles.
template <int NOUT>
__global__ void k_gemm_wmma(const float* __restrict__ X, const float* __restrict__ W,
                            const float* __restrict__ bias, float* __restrict__ H,
                            int n, int ntiles) {
  const int lane = threadIdx.x & 31;
  const int wid = threadIdx.x >> 5;
  const int tile = blockIdx.x * (blockDim.x >> 5) + wid;
  if (tile >= ntiles) return;                 // uniform per-wave exit
  const int m0 = tile * 16;
  const int half = lane >> 4;                 // 0: K lo pair, 1: K hi pair
  const int l = lane & 15;
  constexpr int NT = NOUT / 16;

  v8f acc[NT];
#pragma unroll
  for (int t = 0; t < NT; ++t) acc[t] = (v8f){0,0,0,0,0,0,0,0};

#pragma unroll
  for (int k0 = 0; k0 < 64; k0 += 4) {
    // A fragment: 16x4 f32 -> 2 VGPRs. lanes 0-15: K=k0,k0+1 ; 16-31: K=k0+2,k0+3
    int m = m0 + l;
    if (m >= n) m = n - 1;                    // clamp (stores are guarded)
    const int ka = k0 + half * 2;
    v2f a;
    a.x = X[(size_t)m * 64 + ka];
    a.y = X[(size_t)m * 64 + ka + 1];
#pragma unroll
    for (int t = 0; t < NT; ++t) {
      // B fragment: 4x16 f32 -> 2 VGPRs, mirroring A's K split across halves.
      const int nn = t * 16 + l;
      v2f b;
      b.x = W[(size_t)ka * NOUT + nn];
      b.y = W[(size_t)(ka + 1) * NOUT + nn];
      acc[t] = wmma4(a, b, acc[t]);
    }
  }

  // C/D layout: VGPR v -> rows m0+v (lanes 0-15) and m0+v+8 (lanes 16-31)
#pragma unroll
  for (int t = 0; t < NT; ++t) {
    const int nn = t * 16 + l;
    const float bv = bias[nn];
#pragma unroll
    for (int v = 0; v < 8; ++v) {
      const int m = m0 + v + half * 8;
      if (m < n) H[(size_t)m * NOUT + nn] = acc[t][v] + bv;
    }
  }
}

// ---------------- BatchNorm (training stats) + SELU ----------------
__global__ void k_bn_reduce(const float* __restrict__ H, float* __restrict__ gsum,
                            float* __restrict__ gsq, int n) {
  __shared__ float ssum[256];
  __shared__ float ssq[256];
  const int f = threadIdx.x & 63;
  const int rl = threadIdx.x >> 6;  // 0..3
  float s = 0.0f, q = 0.0f;
  for (int r = blockIdx.x * 4 + rl; r < n; r += gridDim.x * 4) {
    float v = H[(size_t)r * 64 + f];
    s += v;
    q += v * v;
  }
  ssum[threadIdx.x] = s;
  ssq[threadIdx.x] = q;
  __syncthreads();
  if (rl == 0) {
    s = ssum[f] + ssum[64 + f] + ssum[128 + f] + ssum[192 + f];
    q = ssq[f] + ssq[64 + f] + ssq[128 + f] + ssq[192 + f];
    atomicAdd(&gsum[f], s);
    atomicAdd(&gsq[f], q);
  }
}

__global__ void k_bn_stats(const float* __restrict__ gsum, const float* __restrict__ gsq,
                           float* __restrict__ mu, float* __restrict__ rsig, int n) {
  int f = threadIdx.x;
  if (f < 64) {
    float m = gsum[f] / (float)n;
    float var = gsq[f] / (float)n - m * m;
    mu[f] = m;
    rsig[f] = rsqrtf(var + 1e-5f);
  }
}

__global__ void k_bn_selu(float* __restrict__ H, const float* __restrict__ gamma,
                          const float* __restrict__ beta, const float* __restrict__ mu,
                          const float* __restrict__ rsig, long total) {
  const float kScale = 1.0507009873554805f;
  const float kAlpha = 1.6732632423543772f;
  long i = (long)blockIdx.x * blockDim.x + threadIdx.x;
  if (i < total) {
    int f = (int)(i & 63);
    float v = H[i];
    v = gamma[f] * (v - mu[f]) * rsig[f] + beta[f];
    v = (v > 0.0f) ? kScale * v : kScale * kAlpha * (expf(v) - 1.0f);
    H[i] = v;
  }
}

// ---------------- softmax over 32 classes: one 32-lane group per row -------
__global__ void k_softmax32(const float* __restrict__ H, float* __restrict__ out,
                            int n) {
  const int lane = threadIdx.x & 31;
  const int r = (int)(((long)blockIdx.x * blockDim.x + threadIdx.x) >> 5);
  if (r >= n) return;
  float v = H[(size_t)r * 32 + lane];
  float m = v;
#pragma unroll
  for (int off = 16; off > 0; off >>= 1) m = fmaxf(m, __shfl_xor(m, off, 32));
  float e = expf(v - m);
  float s = e;
#pragma unroll
  for (int off = 16; off > 0; off >>= 1) s += __shfl_xor(s, off, 32);
  out[(size_t)r * 32 + lane] = e / s;
}

// ---------------------------------------------------------------------------
extern "C" void kernel_launch(void* const* d_in, const int* in_sizes, int n_in,
                              void* d_out, int out_size, void* d_ws, size_t ws_size,
                              hipStream_t stream) {
  const float* x     = (const float*)d_in[0];
  const int*   row   = (const int*)d_in[1];
  const int*   col   = (const int*)d_in[2];
  const float* w1    = (const float*)d_in[3];
  const float* b1    = (const float*)d_in[4];
  const float* gamma = (const float*)d_in[5];
  const float* beta  = (const float*)d_in[6];
  const float* w2    = (const float*)d_in[7];
  const float* b2    = (const float*)d_in[8];

  const int n = in_sizes[0] / 64;   // 100000
  const int E = in_sizes[1];        // 1600000

  // workspace layout (floats)
  float* ws    = (float*)d_ws;
  float* deg   = ws;
  float* dis   = deg + n;
  float* wedge = dis + n;
  float* buf0  = wedge + E;
  float* buf1  = buf0 + (size_t)n * 64;
  float* h2    = buf1 + (size_t)n * 64;
  float* bnsum = h2 + (size_t)n * 32;
  float* bnsq  = bnsum + 64;
  float* mu    = bnsq + 64;
  float* rsig  = mu + 64;

  // ---- gcn_norm weights ----
  k_zero<<<(n + 255) / 256, 256, 0, stream>>>(deg, n);
  k_deg<<<(E + 255) / 256, 256, 0, stream>>>(row, deg, E);
  k_dis<<<(n + 255) / 256, 256, 0, stream>>>(deg, dis, n);
  k_edgew<<<(E + 255) / 256, 256, 0, stream>>>(row, col, dis, wedge, E);

  const long n64_4 = (long)n * 16;                   // n*64 floats / 4
  dim3 cgrid((unsigned)((n64_4 + 255) / 256));
  dim3 sgrid((unsigned)((E + 15) / 16));

  // ---- conv1 propagation: 4x (out = in + A_norm*in) ----
  const float* cur = x;
  float* bufs[2] = {buf0, buf1};
  for (int s = 0; s < 4; ++s) {
    float* dst = bufs[s & 1];
    k_copy4<<<cgrid, 256, 0, stream>>>((const float4*)cur, (float4*)dst, n64_4);
    k_spmm<<<sgrid, 256, 0, stream>>>(row, col, wedge, cur, dst, E);
    cur = dst;
  }
  float* h1 = (cur == buf0) ? buf1 : buf0;

  // ---- GEMM1 (WMMA fp32): h1 = cur @ w1 + b1 ----
  const int ntiles = (n + 15) / 16;
  k_gemm_wmma<64><<<(ntiles + 7) / 8, 256, 0, stream>>>(cur, w1, b1, h1, n, ntiles);

  // ---- BatchNorm + SELU ----
  k_zero<<<1, 128, 0, stream>>>(bnsum, 128);  // zeros bnsum + bnsq (contiguous)
  k_bn_reduce<<<512, 256, 0, stream>>>(h1, bnsum, bnsq, n);
  k_bn_stats<<<1, 64, 0, stream>>>(bnsum, bnsq, mu, rsig, n);
  k_bn_selu<<<(unsigned)(((long)n * 64 + 255) / 256), 256, 0, stream>>>(
      h1, gamma, beta, mu, rsig, (long)n * 64);

  // ---- conv2 propagation: one step ----
  float* p2 = (h1 == buf0) ? buf1 : buf0;
  k_copy4<<<cgrid, 256, 0, stream>>>((const float4*)h1, (float4*)p2, n64_4);
  k_spmm<<<sgrid, 256, 0, stream>>>(row, col, wedge, h1, p2, E);

  // ---- GEMM2 (WMMA fp32) + softmax ----
  k_gemm_wmma<32><<<(ntiles + 7) / 8, 256, 0, stream>>>(p2, w2, b2, h2, n, ntiles);
  k_softmax32<<<(n + 7) / 8, 256, 0, stream>>>(h2, (float*)d_out, n);
}